// MessagePassing_82875688943834
// MI455X (gfx1250) — compile-verified
//
#include <hip/hip_runtime.h>

typedef float v2f __attribute__((ext_vector_type(2)));
typedef float v8f __attribute__((ext_vector_type(8)));

#define N_NODES 50000
#define N_EDGES 800000
#define DIM     128
#define DOUT    64

// ---------------------------------------------------------------------------
// zero a float buffer
__global__ void __launch_bounds__(256) zero_f32(float* __restrict__ p, long long n) {
    long long i = (long long)blockIdx.x * 256 + threadIdx.x;
    if (i < n) p[i] = 0.0f;
}

// count out-degree per src node (float atomics, feeds 1/max(cnt,1))
__global__ void __launch_bounds__(256) count_deg(const long long* __restrict__ edges,
                                                 float* __restrict__ cnt, int nE) {
    int e = blockIdx.x * 256 + threadIdx.x;
    if (e < nE) {
        int s = (int)edges[2 * e];
        atomicAdd(&cnt[s], 1.0f);
    }
}

__global__ void __launch_bounds__(256) finalize_rdenom(float* __restrict__ cnt, int n) {
    int i = blockIdx.x * 256 + threadIdx.x;
    if (i < n) cnt[i] = 1.0f / fmaxf(cnt[i], 1.0f);
}

// gather h[dst] and atomically accumulate into agg[src]; one thread = one
// (edge, 4-float chunk). 128-bit loads, 4x global_atomic_add_f32.
__global__ void __launch_bounds__(256) scatter_add(const long long* __restrict__ edges,
                                                   const float* __restrict__ h,
                                                   float* __restrict__ agg,
                                                   long long nwork) {
    long long idx = (long long)blockIdx.x * 256 + threadIdx.x;
    if (idx >= nwork) return;              // nwork = N_EDGES * (DIM/4)
    int e  = (int)(idx >> 5);              // DIM/4 == 32 chunks per edge
    int f4 = ((int)idx & 31) << 2;
    int s = (int)edges[2 * e];
    int d = (int)edges[2 * e + 1];
    const float4 v = *(const float4*)(h + (size_t)d * DIM + f4);
    float* ap = agg + (size_t)s * DIM + f4;
    atomicAdd(ap + 0, v.x);
    atomicAdd(ap + 1, v.y);
    atomicAdd(ap + 2, v.z);
    atomicAdd(ap + 3, v.w);
}

// ---------------------------------------------------------------------------
// Fused WMMA GEMM (full fp32 matrix pipe, V_WMMA_F32_16X16X4_F32):
//   C[M,N] = act( A1[M,K] @ B1[K,N]  (+ (rscale[m]*A2[M,K]) @ B2[K,N])  + bias[N] )
// One wave computes one 16x16 C tile; K stepped by 4 per WMMA.
//
// Lane layouts per CDNA5 ISA 7.12.2 (32-bit matrices, wave32):
//   A 16x4 : lanes 0-15 -> M=lane,   VGPR0=K0, VGPR1=K1
//            lanes16-31 -> M=lane-16,VGPR0=K2, VGPR1=K3
//   B 4x16 : VGPR0: lanes0-15 row K0, lanes16-31 row K2 ; VGPR1: K1 / K3 (N=lane&15)
//   C 16x16: VGPR v: lanes0-15 -> M=v, lanes16-31 -> M=v+8 ; N=lane&15
__global__ void __launch_bounds__(256) wmma_gemm_f32(
        const float* __restrict__ A1, const float* __restrict__ B1,
        const float* __restrict__ A2, const float* __restrict__ B2,
        const float* __restrict__ rscale,
        const float* __restrict__ bias, float* __restrict__ C,
        int M, int N, int K, int relu)
{
    const int lane   = threadIdx.x & 31;
    const int widb   = threadIdx.x >> 5;
    const int ntiles = N >> 4;
    const int mtiles = M >> 4;
    const int wave   = blockIdx.x * (blockDim.x >> 5) + widb;
    const int mt = wave / ntiles;
    const int nt = wave % ntiles;
    if (mt >= mtiles) return;              // whole-wave uniform exit: EXEC stays all-1s for WMMA

    const int half = lane >> 4;            // 0: K0/K1 half, 1: K2/K3 half
    const int lm   = lane & 15;
    const int arow = mt * 16 + lm;         // A row this lane feeds
    const int col  = nt * 16 + lm;         // C column this lane owns

    v8f acc = {};
    const float* a1p = A1 + (size_t)arow * K;
    for (int k = 0; k < K; k += 4) {
        v2f a, b;
        a.x = a1p[k + 2 * half];
        a.y = a1p[k + 2 * half + 1];
        b.x = B1[(size_t)(k + 2 * half)     * N + col];
        b.y = B1[(size_t)(k + 2 * half + 1) * N + col];
        acc = __builtin_amdgcn_wmma_f32_16x16x4_f32(false, a, false, b,
                                                    (short)0, acc, false, false);
    }
    if (A2 != nullptr) {
        const float rs = rscale ? rscale[arow] : 1.0f;   // fold 1/denom into agg loads
        const float* a2p = A2 + (size_t)arow * K;
        for (int k = 0; k < K; k += 4) {
            v2f a, b;
            a.x = a2p[k + 2 * half] * rs;
            a.y = a2p[k + 2 * half + 1] * rs;
            b.x = B2[(size_t)(k + 2 * half)     * N + col];
            b.y = B2[(size_t)(k + 2 * half + 1) * N + col];
            acc = __builtin_amdgcn_wmma_f32_16x16x4_f32(false, a, false, b,
                                                        (short)0, acc, false, false);
        }
    }

    const float bv = bias[col];
    #pragma unroll
    for (int v = 0; v < 8; ++v) {
        int r = mt * 16 + v + 8 * half;
        float val = acc[v] + bv;
        if (relu) val = fmaxf(val, 0.0f);
        C[(size_t)r * N + col] = val;
    }
}

// ---------------------------------------------------------------------------
extern "C" void kernel_launch(void* const* d_in, const int* in_sizes, int n_in,
                              void* d_out, int out_size, void* d_ws, size_t ws_size,
                              hipStream_t stream) {
    const float*     features = (const float*)d_in[0];
    const long long* edges    = (const long long*)d_in[1];   // int64 per reference
    const float*     W_enc    = (const float*)d_in[2];
    const float*     b_enc    = (const float*)d_in[3];
    const float*     W_self   = (const float*)d_in[4];       // [2,128,128]
    const float*     W_neigh  = (const float*)d_in[5];       // [2,128,128]
    const float*     b_comb   = (const float*)d_in[6];       // [2,128]
    const float*     W_out    = (const float*)d_in[7];       // [128,64]
    const float*     b_out    = (const float*)d_in[8];
    float* out = (float*)d_out;

    const long long NH = (long long)N_NODES * DIM;           // 6.4M floats
    float* hA     = (float*)d_ws;
    float* hB     = hA + NH;
    float* agg    = hB + NH;
    float* rdenom = agg + NH;                                // 50000 floats

    auto cdiv = [](long long a, long long b) { return (int)((a + b - 1) / b); };

    // --- degree / reciprocal denominator --------------------------------
    zero_f32<<<cdiv(N_NODES, 256), 256, 0, stream>>>(rdenom, N_NODES);
    count_deg<<<cdiv(N_EDGES, 256), 256, 0, stream>>>(edges, rdenom, N_EDGES);
    finalize_rdenom<<<cdiv(N_NODES, 256), 256, 0, stream>>>(rdenom, N_NODES);

    // --- encoder: hA = features @ W_enc + b_enc -------------------------
    {
        int waves = (N_NODES / 16) * (DIM / 16);             // 3125 * 8
        wmma_gemm_f32<<<cdiv(waves, 8), 256, 0, stream>>>(
            features, W_enc, nullptr, nullptr, nullptr, b_enc, hA,
            N_NODES, DIM, DIM, 0);
    }

    // --- 2 message-passing layers ---------------------------------------
    const long long scatter_work = (long long)N_EDGES * (DIM / 4);
    const float* h_in  = hA;
    float*       h_out = hB;
    for (int l = 0; l < 2; ++l) {
        zero_f32<<<cdiv(NH, 256), 256, 0, stream>>>(agg, NH);
        scatter_add<<<cdiv(scatter_work, 256), 256, 0, stream>>>(edges, h_in, agg, scatter_work);
        int waves = (N_NODES / 16) * (DIM / 16);
        wmma_gemm_f32<<<cdiv(waves, 8), 256, 0, stream>>>(
            h_in, W_self + (size_t)l * DIM * DIM,
            agg,  W_neigh + (size_t)l * DIM * DIM,
            rdenom, b_comb + (size_t)l * DIM, h_out,
            N_NODES, DIM, DIM, 1);
        // swap ping-pong
        const float* t = h_in; h_in = h_out; h_out = (float*)t;
    }

    // --- output head: out = h @ W_out + b_out ---------------------------
    {
        int waves = (N_NODES / 16) * (DOUT / 16);            // 3125 * 4
        wmma_gemm_f32<<<cdiv(waves, 8), 256, 0, stream>>>(
            h_in, W_out, nullptr, nullptr, nullptr, b_out, out,
            N_NODES, DOUT, DIM, 0);
    }
}